// _ProposalLayer_FPN_45286135169247
// MI455X (gfx1250) — compile-verified
//
#include <hip/hip_runtime.h>

// ---------------------------------------------------------------------------
// ProposalLayer FPN for MI455X (gfx1250, wave32)
//   scores:      (32, 49104, 1) f32
//   bbox_deltas: (32, 49104, 4) f32
//   out:         (32, 300, 5)   f32  [batch_idx, x1, y1, x2, y2]
// ---------------------------------------------------------------------------

#define NB      32
#define NANCH   49104
#define PRE_N   6000
#define POST_N  300
#define NMS_TH  0.7f
#define IMGF    512.0f

typedef __attribute__((ext_vector_type(2))) float v2f;
typedef __attribute__((ext_vector_type(8))) float v8f;
typedef int v4i __attribute__((vector_size(16)));   // matches builtin's vector pointee

struct Ctl {
  unsigned T;      // exact threshold bit pattern (6000th score)
  int above;       // count of scores strictly > T
  int r;           // how many ==T elements to take (above + r == 6000)
  int cntA;        // atomic slot counter for > T
  int cntB;        // atomic slot counter for == T
  int pad0, pad1, pad2;
};

// ---- CDNA5 async global->LDS copy (ASYNCcnt path) -------------------------

__device__ __forceinline__ void async_to_lds_b32(const void* g, void* l) {
#if __has_builtin(__builtin_amdgcn_global_load_async_to_lds_b32)
  __builtin_amdgcn_global_load_async_to_lds_b32(
      (__attribute__((address_space(1))) int*)(unsigned long long)(size_t)g,
      (__attribute__((address_space(3))) int*)(unsigned)(size_t)l, 0, 0);
#else
  unsigned lo = (unsigned)(size_t)l;  // low 32 bits of shared-aperture addr == LDS offset
  asm volatile("global_load_async_to_lds_b32 %0, %1, off"
               :: "v"(lo), "v"(g) : "memory");
#endif
}

__device__ __forceinline__ void async_to_lds_b128(const void* g, void* l) {
#if __has_builtin(__builtin_amdgcn_global_load_async_to_lds_b128)
  __builtin_amdgcn_global_load_async_to_lds_b128(
      (__attribute__((address_space(1))) v4i*)(unsigned long long)(size_t)g,
      (__attribute__((address_space(3))) v4i*)(unsigned)(size_t)l, 0, 0);
#else
  unsigned lo = (unsigned)(size_t)l;
  asm volatile("global_load_async_to_lds_b128 %0, %1, off"
               :: "v"(lo), "v"(g) : "memory");
#endif
}

__device__ __forceinline__ void wait_asynccnt0() {
#if __has_builtin(__builtin_amdgcn_s_wait_asynccnt)
  __builtin_amdgcn_s_wait_asynccnt(0);
#else
  asm volatile("s_wait_asynccnt 0x0" ::: "memory");
#endif
}

// ---------------------------------------------------------------------------
// Kernel 1: exact radix-select of the 6000th largest score per batch.
// One 1024-thread block per batch. 3 passes over float bit fields
// [31:20], [19:8], [7:0]; histogram tree reduced with V_WMMA_F32_16X16X4_F32.
// ---------------------------------------------------------------------------
__global__ __launch_bounds__(1024) void rpn_select(const float* __restrict__ scores,
                                                   Ctl* __restrict__ ctl)
{
  __shared__ unsigned hist[4096];
  __shared__ float s4[256];     // sums of 16 bins each
  __shared__ float hsum[64];    // sums of 64 bins each (via WMMA)
  __shared__ unsigned s_pref, s_mask;
  __shared__ int s_K, s_above;

  const int b   = blockIdx.x;
  const int tid = threadIdx.x;
  const float* sptr = scores + (size_t)b * NANCH;

  if (tid == 0) { s_pref = 0u; s_mask = 0u; s_K = PRE_N; s_above = 0; }
  __syncthreads();

  const int      shifts[3] = {20, 8, 0};
  const unsigned kbits [3] = {0xFFFu, 0xFFFu, 0xFFu};

  for (int pass = 0; pass < 3; ++pass) {
    for (int i = tid; i < 4096; i += 1024) hist[i] = 0u;
    __syncthreads();

    const unsigned pref = s_pref, mask = s_mask;
    const int      sh   = shifts[pass];
    const unsigned km   = kbits[pass];

    for (int i = tid; i < NANCH; i += 1024) {
      unsigned v = __float_as_uint(sptr[i]);   // scores in [0,1): bits monotone
      if ((v & mask) == pref) {
        unsigned key = (v >> sh) & km;
        atomicAdd(&hist[key], 1u);
      }
    }
    __syncthreads();

    if (tid < 256) {
      float s = 0.f;
      #pragma unroll
      for (int t = 0; t < 16; ++t) s += (float)hist[16 * tid + t];
      s4[tid] = s;
    }
    __syncthreads();

    // 256 partials -> 64 sums-of-4 using the matrix engine: D = A(16x4) * ones(4x16)
#if __has_builtin(__builtin_amdgcn_wmma_f32_16x16x4_f32)
    if (tid < 32) {              // wave 0 only, EXEC all-ones
      const int lane = tid;
      const int m    = lane & 15;
      const int koff = (lane >> 4) ? 2 : 0;   // A 16x4 f32 layout (ISA 7.12.2)
      for (int blk = 0; blk < 4; ++blk) {
        v2f a;
        a.x = s4[blk * 64 + 4 * m + koff];
        a.y = s4[blk * 64 + 4 * m + koff + 1];
        v2f bones; bones.x = 1.0f; bones.y = 1.0f;   // ones: layout independent
        v8f cz = {0.f, 0.f, 0.f, 0.f, 0.f, 0.f, 0.f, 0.f};
        v8f d = __builtin_amdgcn_wmma_f32_16x16x4_f32(
            false, a, false, bones, (short)0, cz, false, false);
        // D[m][n] == rowsum(m); C/D layout: lane<16 -> rows v, lane>=16 -> rows v+8
        if (lane == 0) {
          #pragma unroll
          for (int v = 0; v < 8; ++v) hsum[blk * 16 + v] = d[v];
        } else if (lane == 16) {
          #pragma unroll
          for (int v = 0; v < 8; ++v) hsum[blk * 16 + 8 + v] = d[v];
        }
      }
    }
#else
    if (tid < 64) {
      float s = 0.f;
      #pragma unroll
      for (int t = 0; t < 4; ++t) s += s4[4 * tid + t];
      hsum[tid] = s;
    }
#endif
    __syncthreads();

    if (tid == 0) {
      int K = s_K;
      int acc = 0;
      int cj = 63;
      for (; cj > 0; --cj) {                 // coarse scan from the top
        int hv = (int)(hsum[cj] + 0.5f);
        if (acc + hv >= K) break;
        acc += hv;
      }
      int bf = 63;
      for (; bf > 0; --bf) {                 // fine scan inside coarse block
        int cv = (int)hist[(cj << 6) + bf];
        if (acc + cv >= K) break;
        acc += cv;
      }
      unsigned sel = (unsigned)((cj << 6) + bf);
      s_pref  = s_pref | (sel << sh);
      s_mask  = s_mask | (km << sh);
      s_above += acc;
      s_K = K - acc;                         // still needed from the boundary bin
    }
    __syncthreads();
  }

  if (tid == 0) {
    Ctl c;
    c.T = s_pref; c.above = s_above; c.r = s_K;
    c.cntA = 0; c.cntB = 0; c.pad0 = c.pad1 = c.pad2 = 0;
    ctl[b] = c;
  }
}

// ---------------------------------------------------------------------------
// Kernel 2: anchor synthesis + box decode + clip + threshold compaction.
// ---------------------------------------------------------------------------
__global__ __launch_bounds__(256) void rpn_compact(const float*  __restrict__ scores,
                                                   const float4* __restrict__ deltas,
                                                   Ctl*          __restrict__ ctl,
                                                   float*        __restrict__ csc,
                                                   float4*       __restrict__ cbx)
{
  const int b = blockIdx.y;
  const int i = blockIdx.x * 256 + threadIdx.x;
  if (i >= NANCH) return;

  const float sval = scores[(size_t)b * NANCH + i];
  const unsigned v = __float_as_uint(sval);
  const unsigned T = ctl[b].T;
  if (v < T) return;

  int slot;
  if (v > T) {
    slot = atomicAdd(&ctl[b].cntA, 1);
  } else {
    int t = atomicAdd(&ctl[b].cntB, 1);
    if (t >= ctl[b].r) return;               // only take r of the ==T ties
    slot = ctl[b].above + t;
  }
  if (slot >= PRE_N) return;                 // safety clamp

  // ---- anchor from flat index (levels 3..7, 9 anchors per cell) ----
  int off, fs, stride;
  if      (i < 36864) { off = 0;     fs = 64; stride = 8;   }
  else if (i < 46080) { off = 36864; fs = 32; stride = 16;  }
  else if (i < 48384) { off = 46080; fs = 16; stride = 32;  }
  else if (i < 48960) { off = 48384; fs = 8;  stride = 64;  }
  else                { off = 48960; fs = 4;  stride = 128; }
  const int j    = i - off;
  const int k    = j % 9;
  const int cell = j / 9;
  const int xg   = cell % fs;
  const int yg   = cell / fs;

  const int   si    = k % 3;
  const int   ri    = k / 3;
  const float size  = (float)(stride * 4);                 // 2^(l+2)
  const float scale = (si == 0) ? 1.0f
                    : (si == 1) ? 1.2599210498948732f      // 2^(1/3)
                                : 1.5874010519681994f;     // 2^(2/3)
  const float sqr   = (ri == 0) ? 0.70710678118654752f
                    : (ri == 1) ? 1.0f
                                : 1.41421356237309505f;    // sqrt(ratio)
  const float aw = size * scale / sqr;
  const float ah = size * scale * sqr;
  const float cx = (xg + 0.5f) * (float)stride;
  const float cy = (yg + 0.5f) * (float)stride;

  // ---- decode + clip ----
  const float4 dd = deltas[(size_t)b * NANCH + i];
  const float dx = dd.x * 0.1f, dy = dd.y * 0.1f;
  const float dw = dd.z * 0.2f, dh = dd.w * 0.2f;
  const float pcx = cx + dx * aw;
  const float pcy = cy + dy * ah;
  const float pw  = expf(dw) * aw;
  const float ph  = expf(dh) * ah;
  const float x1 = fminf(fmaxf(pcx - 0.5f * pw, 0.f), IMGF);
  const float y1 = fminf(fmaxf(pcy - 0.5f * ph, 0.f), IMGF);
  const float x2 = fminf(fmaxf(pcx + 0.5f * pw, 0.f), IMGF);
  const float y2 = fminf(fmaxf(pcy + 0.5f * ph, 0.f), IMGF);

  csc[(size_t)b * PRE_N + slot] = sval;
  cbx[(size_t)b * PRE_N + slot] = make_float4(x1, y1, x2, y2);
}

// ---------------------------------------------------------------------------
// Kernel 3: greedy NMS, one block per batch, full working set in LDS (144 KB
// of the 320 KB WGP LDS). Candidates staged with async global->LDS loads.
// ---------------------------------------------------------------------------
__global__ __launch_bounds__(1024) void rpn_nms(const float*  __restrict__ csc,
                                                const float4* __restrict__ cbx,
                                                float*        __restrict__ out)
{
  extern __shared__ char smem[];
  float4* bx = (float4*)(smem);             // 6000 * 16 = 96000 B
  float*  sc = (float* )(smem + 96000);     // 6000 * 4  = 24000 B
  float*  ar = (float* )(smem + 120000);    // 6000 * 4  = 24000 B
  float*  wv = (float* )(smem + 144000);    // 32 wave maxima
  int*    wi = (int*  )(smem + 144128);     // 32 wave argmaxes
  float*  bestv = (float*)(smem + 144256);
  int*    besti = (int*  )(smem + 144260);

  const int b    = blockIdx.x;
  const int tid  = threadIdx.x;
  const int lane = tid & 31;                // wave32
  const int wave = tid >> 5;

  const float4* gbx = cbx + (size_t)b * PRE_N;
  const float*  gsc = csc + (size_t)b * PRE_N;

  // async stage: boxes (b128) and scores (b32) into LDS
  for (int j = tid; j < PRE_N; j += 1024) async_to_lds_b128(gbx + j, bx + j);
  for (int j = tid; j < PRE_N; j += 1024) async_to_lds_b32 (gsc + j, sc + j);
  wait_asynccnt0();
  __syncthreads();

  for (int j = tid; j < PRE_N; j += 1024) {
    float4 p = bx[j];
    ar[j] = (p.z - p.x) * (p.w - p.y);
  }
  __syncthreads();

  const float NEG = -__builtin_inff();
  float* orow = out + (size_t)b * POST_N * 5;

  for (int it = 0; it < POST_N; ++it) {
    // ---- block argmax over LDS scores ----
    float bvL = NEG; int biL = -1;
    for (int j = tid; j < PRE_N; j += 1024) {
      float s = sc[j];
      if (s > bvL) { bvL = s; biL = j; }
    }
    #pragma unroll
    for (int off = 16; off > 0; off >>= 1) {
      float ov = __shfl_down(bvL, off, 32);
      int   oi = __shfl_down(biL, off, 32);
      if (ov > bvL) { bvL = ov; biL = oi; }
    }
    if (lane == 0) { wv[wave] = bvL; wi[wave] = biL; }
    __syncthreads();
    if (wave == 0) {
      float v2 = wv[lane]; int i2 = wi[lane];
      #pragma unroll
      for (int off = 16; off > 0; off >>= 1) {
        float ov = __shfl_down(v2, off, 32);
        int   oi = __shfl_down(i2, off, 32);
        if (ov > v2) { v2 = ov; i2 = oi; }
      }
      if (lane == 0) { *bestv = v2; *besti = i2; }
    }
    __syncthreads();

    const float bvv = *bestv;
    const int   bj  = *besti;
    const bool  valid = (bvv > NEG) && (bj >= 0);
    float4 bb; float arB = 0.f;
    if (valid) { bb = bx[bj]; arB = ar[bj]; }
    else       { bb.x = bb.y = bb.z = bb.w = 0.f; }

    if (tid == 0) {
      orow[it * 5 + 0] = (float)b;
      orow[it * 5 + 1] = bb.x;
      orow[it * 5 + 2] = bb.y;
      orow[it * 5 + 3] = bb.z;
      orow[it * 5 + 4] = bb.w;
    }

    if (valid) {
      for (int j = tid; j < PRE_N; j += 1024) {
        float s = sc[j];
        if (s == NEG) continue;
        float4 p = bx[j];
        float xx1 = fmaxf(p.x, bb.x);
        float yy1 = fmaxf(p.y, bb.y);
        float xx2 = fminf(p.z, bb.z);
        float yy2 = fminf(p.w, bb.w);
        float inter = fmaxf(xx2 - xx1, 0.f) * fmaxf(yy2 - yy1, 0.f);
        float iou = inter / (ar[j] + arB - inter + 1e-8f);
        if (iou > NMS_TH || j == bj) sc[j] = NEG;
      }
    }
    __syncthreads();
  }
}

// ---------------------------------------------------------------------------
extern "C" void kernel_launch(void* const* d_in, const int* in_sizes, int n_in,
                              void* d_out, int out_size, void* d_ws, size_t ws_size,
                              hipStream_t stream)
{
  const float*  scores = (const float* )d_in[0];   // (32, 49104, 1) f32
  const float4* deltas = (const float4*)d_in[1];   // (32, 49104, 4) f32

  unsigned char* ws = (unsigned char*)d_ws;
  Ctl*    ctl = (Ctl*)ws;                                             // 1024 B
  float*  csc = (float*)(ws + 1024);                                  // 32*6000 f32
  float4* cbx = (float4*)(ws + 1024 + (size_t)NB * PRE_N * sizeof(float)); // 16B aligned

  rpn_select<<<NB, 1024, 0, stream>>>(scores, ctl);

  dim3 g2((NANCH + 255) / 256, NB);
  rpn_compact<<<g2, 256, 0, stream>>>(scores, deltas, ctl, csc, cbx);

  const size_t shmem = 144512;   // boxes + scores + areas + reduction scratch
  rpn_nms<<<NB, 1024, shmem, stream>>>(csc, cbx, (float*)d_out);
}